// EdgesToGlobalsAggregator_83631603188038
// MI455X (gfx1250) — compile-verified
//
#include <hip/hip_runtime.h>

// Segment-sum of edge features into per-graph rows.
//   edges : [total_edges, 128] fp32 (contiguous per graph)
//   n_edge: [num_graphs] int32
//   out   : [num_graphs, 128] fp32
//
// Strategy (MI455X): memory-bound streaming reduction (~1.07 GB read,
// ~46us floor at 23.3 TB/s). Reduction is folded into the matrix pipe via
// V_WMMA_F32_16X16X4_F32 with an all-ones A matrix:
//   D[m][n] = sum_k B[k][n]  (+ C accumulate)
// Because A is all-ones, only the N (feature) lane position of B matters,
// making the construction robust to the exact K slot mapping.

typedef float v2f __attribute__((ext_vector_type(2)));
typedef float v8f __attribute__((ext_vector_type(8)));

#define D_FEAT 128

__global__ void seg_offsets_kernel(const int* __restrict__ n_edge, int num_graphs,
                                   int* __restrict__ offsets) {
  // Tiny (1024-element) exclusive prefix sum; serial is fine.
  if (blockIdx.x == 0 && threadIdx.x == 0) {
    int acc = 0;
    for (int g = 0; g < num_graphs; ++g) {
      offsets[g] = acc;
      acc += n_edge[g];
    }
    offsets[num_graphs] = acc;
  }
}

__global__ __launch_bounds__(256) void
seg_sum_wmma_kernel(const float* __restrict__ edges,
                    const int* __restrict__ offsets,
                    float* __restrict__ out) {
  const int g     = blockIdx.x;
  const int begin = offsets[g];
  const int count = offsets[g + 1] - begin;

  const int wave = threadIdx.x >> 5;   // 8 waves: feature slices of 16
  const int lane = threadIdx.x & 31;
  const int f0   = wave << 4;
  const int n    = lane & 15;          // B/D matrix N position == lane&15
  const int kh   = lane >> 4;          // which lane-half (K slot selector)

  // Per-lane column base: feature (f0 + n) of the first edge row of graph g.
  const float* base = edges + (size_t)begin * D_FEAT + f0 + n;

  const v2f a = {1.0f, 1.0f};          // all-ones A (16x4): D = colsum(B) + C
  v8f c0 = {}; v8f c1 = {}; v8f c2 = {}; v8f c3 = {};

  const int full = count & ~15;        // 16 edge rows per iteration per wave
  for (int e = 0; e < full; e += 16) {
    const float* p = base + (size_t)e * D_FEAT;
    __builtin_prefetch(p + 16 * D_FEAT, 0, 3);   // global_prefetch_b8, 8KB ahead

    // Each B operand covers 4 edge rows x 16 features (K mapping is
    // irrelevant since A is all-ones; N mapping is lane&15, already in base).
    v2f b0 = { p[(size_t)(0  + kh) * D_FEAT], p[(size_t)(2  + kh) * D_FEAT] };
    v2f b1 = { p[(size_t)(4  + kh) * D_FEAT], p[(size_t)(6  + kh) * D_FEAT] };
    v2f b2 = { p[(size_t)(8  + kh) * D_FEAT], p[(size_t)(10 + kh) * D_FEAT] };
    v2f b3 = { p[(size_t)(12 + kh) * D_FEAT], p[(size_t)(14 + kh) * D_FEAT] };

    c0 = __builtin_amdgcn_wmma_f32_16x16x4_f32(false, a, false, b0, (short)0, c0, false, false);
    c1 = __builtin_amdgcn_wmma_f32_16x16x4_f32(false, a, false, b1, (short)0, c1, false, false);
    c2 = __builtin_amdgcn_wmma_f32_16x16x4_f32(false, a, false, b2, (short)0, c2, false, false);
    c3 = __builtin_amdgcn_wmma_f32_16x16x4_f32(false, a, false, b3, (short)0, c3, false, false);
  }

  // Tail (<16 rows): zero-fill missing rows with predicated loads so that
  // EXEC remains all-ones at the WMMA itself.
  if (full < count) {
    const float* p   = base + (size_t)full * D_FEAT;
    const int    rem = count - full;
#define LDZ(r) (((r) < rem) ? p[(size_t)(r) * D_FEAT] : 0.0f)
    v2f b0 = { LDZ(0 + kh),  LDZ(2 + kh)  };
    v2f b1 = { LDZ(4 + kh),  LDZ(6 + kh)  };
    v2f b2 = { LDZ(8 + kh),  LDZ(10 + kh) };
    v2f b3 = { LDZ(12 + kh), LDZ(14 + kh) };
#undef LDZ
    c0 = __builtin_amdgcn_wmma_f32_16x16x4_f32(false, a, false, b0, (short)0, c0, false, false);
    c1 = __builtin_amdgcn_wmma_f32_16x16x4_f32(false, a, false, b1, (short)0, c1, false, false);
    c2 = __builtin_amdgcn_wmma_f32_16x16x4_f32(false, a, false, b2, (short)0, c2, false, false);
    c3 = __builtin_amdgcn_wmma_f32_16x16x4_f32(false, a, false, b3, (short)0, c3, false, false);
  }

  // Every D row/VGPR holds the identical column sums (A was all-ones):
  // element 0, lanes 0..15 carry out[f0 + n]. Combine the 4 accumulators.
  float s = (c0[0] + c1[0]) + (c2[0] + c3[0]);
  if (lane < 16) {
    out[(size_t)g * D_FEAT + f0 + lane] = s;
  }
}

extern "C" void kernel_launch(void* const* d_in, const int* in_sizes, int n_in,
                              void* d_out, int out_size, void* d_ws, size_t ws_size,
                              hipStream_t stream) {
  const float* edges  = (const float*)d_in[0];
  const int*   n_edge = (const int*)d_in[1];
  const int num_graphs = in_sizes[1];

  int* offsets = (int*)d_ws;  // (num_graphs + 1) ints of scratch

  seg_offsets_kernel<<<1, 32, 0, stream>>>(n_edge, num_graphs, offsets);
  seg_sum_wmma_kernel<<<num_graphs, 256, 0, stream>>>(edges, offsets, (float*)d_out);
}